// InsGNBNIN_78237124264115
// MI455X (gfx1250) — compile-verified
//
#include <hip/hip_runtime.h>

#define N_IMG 8
#define C_CH  64
#define HW_SZ 65536
#define K_INST 16
#define EPSF 1e-5f

typedef float v2f __attribute__((ext_vector_type(2)));
typedef float v4f __attribute__((ext_vector_type(4)));
typedef float v8f __attribute__((ext_vector_type(8)));

// ---------------- zero stats ----------------
__global__ void zero_k(float* sum, float* ssq, float* cnt) {
    int i = blockIdx.x * 256 + threadIdx.x;
    if (i < N_IMG * K_INST * C_CH) { sum[i] = 0.f; ssq[i] = 0.f; }
    if (i < N_IMG * K_INST) cnt[i] = 0.f;
}

// ---------------- prep: int32 ids -> int8, per-(n,k) counts ----------------
__global__ __launch_bounds__(256) void prep_k(const int* __restrict__ ids,
                                              signed char* __restrict__ id8,
                                              float* __restrict__ cnt) {
    __shared__ int hist[K_INST];
    int tid = threadIdx.x;
    if (tid < K_INST) hist[tid] = 0;
    __syncthreads();
    int base = blockIdx.x * 1024;          // 1024 ids per block, stays in one image
    int n = base >> 16;                    // HW = 65536
    #pragma unroll
    for (int k = 0; k < 4; ++k) {
        int i = base + tid + k * 256;
        int id = ids[i];
        signed char v = -1;
        if (id >= 0) {
            int kk = id < (K_INST - 1) ? id : (K_INST - 1);   // clip like reference
            v = (signed char)kk;
            atomicAdd(&hist[kk], 1);
        }
        id8[i] = v;
    }
    __syncthreads();
    if (tid < K_INST) atomicAdd(&cnt[n * K_INST + tid], (float)hist[tid]);
}

// ---------------- stats: one-hot x WMMA segmented reduction ----------------
// grid: 512 blocks = (n:8) x (cTile:4 of 16 channels) x (chunk:16 of 4096 px)
// Double-buffered LDS tiles: 128 pixels x 16 channels, transposed [p][c] pad 17.
// A = onehot(ids) 16x4 (M=instance, K=pixel), B = x 4x16 (K=pixel, N=channel).
// D accumulates [instance x channel] sums; second WMMA with B*B -> sum of squares.
__global__ __launch_bounds__(256) void stats_wmma(const float* __restrict__ x,
                                                  const signed char* __restrict__ id8,
                                                  float* __restrict__ sum,
                                                  float* __restrict__ ssq) {
    __shared__ float xT[2][128 * 17];
    __shared__ int   idT[2][32];
    __shared__ float sred[512];            // [0..255]=sum, [256..511]=ssq (16x16 each)

    int b = blockIdx.x;
    int chunk = b & 15, cTile = (b >> 4) & 3, n = b >> 6;
    int cBase = cTile * 16;
    int tid  = threadIdx.x;
    int lane = tid & 31;
    int w    = tid >> 5;                   // wave id, 8 waves
    int hi   = lane >> 4;                  // lane half selects K{0,1} vs K{2,3}
    int m    = lane & 15;                  // instance row (A) / channel col (B,D)

    sred[tid] = 0.f; sred[tid + 256] = 0.f;

    // staging assignment: 512 float4s per tile; thread handles f0 = tid, f1 = tid+256
    int c0 = tid >> 5,        p40 = tid & 31;
    int c1 = (tid + 256) >> 5, p41 = tid & 31;   // (tid+256)&31 == tid&31
    int sb0 = p40 * 68 + c0;               // (p4*4)*17 + c
    int sb1 = p41 * 68 + c1;

    v8f dsum = {}; v8f dssq = {};
    const float* xp = x + ((size_t)(n * C_CH + cBase) << 16);
    const signed char* idp = id8 + n * HW_SZ;
    const v4f* src0 = (const v4f*)(xp + ((size_t)c0 << 16));
    const v4f* src1 = (const v4f*)(xp + ((size_t)c1 << 16));

    // prologue: stage tile 0 into buffer 0
    int pBase = chunk * 4096;
    v4f r0 = src0[(pBase >> 2) + p40];
    v4f r1 = src1[(pBase >> 2) + p41];
    int idv = 0;
    if (tid < 32) idv = ((const int*)(idp + pBase))[tid];
    {
        float* dst = xT[0];
        dst[sb0] = r0[0]; dst[sb0 + 17] = r0[1]; dst[sb0 + 34] = r0[2]; dst[sb0 + 51] = r0[3];
        dst[sb1] = r1[0]; dst[sb1 + 17] = r1[1]; dst[sb1 + 34] = r1[2]; dst[sb1 + 51] = r1[3];
        if (tid < 32) idT[0][tid] = idv;
    }

    for (int st = 0; st < 32; ++st) {
        __syncthreads();                   // buffer (st&1) fully staged for all waves
        int cur = st & 1, nxt = cur ^ 1;
        if (st < 31) {                     // issue next tile's global loads early
            int pB2 = chunk * 4096 + (st + 1) * 128;
            r0 = src0[(pB2 >> 2) + p40];
            r1 = src1[(pB2 >> 2) + p41];
            if (tid < 32) idv = ((const int*)(idp + pB2))[tid];
        }
        const signed char* idb = (const signed char*)idT[cur];
        const float* xb = xT[cur];
        // wave w handles pixels [w*16, w*16+16): 4 WMMA K-steps of 4 pixels
        #pragma unroll
        for (int s = 0; s < 4; ++s) {
            int q   = w * 16 + s * 4;
            int pk0 = q + hi * 2;          // A/B f32 layout: vgpr0 = K0|K2, vgpr1 = K1|K3
            int pk1 = pk0 + 1;
            float a0 = (idb[pk0] == m) ? 1.f : 0.f;
            float a1 = (idb[pk1] == m) ? 1.f : 0.f;
            float b0 = xb[pk0 * 17 + m];
            float b1 = xb[pk1 * 17 + m];
            v2f av = {a0, a1};
            v2f bv = {b0, b1};
            v2f b2 = {b0 * b0, b1 * b1};
            dsum = __builtin_amdgcn_wmma_f32_16x16x4_f32(false, av, false, bv,
                                                         (short)0, dsum, false, false);
            dssq = __builtin_amdgcn_wmma_f32_16x16x4_f32(false, av, false, b2,
                                                         (short)0, dssq, false, false);
        }
        if (st < 31) {                     // stage next tile into the other buffer
            float* dst = xT[nxt];
            dst[sb0] = r0[0]; dst[sb0 + 17] = r0[1]; dst[sb0 + 34] = r0[2]; dst[sb0 + 51] = r0[3];
            dst[sb1] = r1[0]; dst[sb1 + 17] = r1[1]; dst[sb1 + 34] = r1[2]; dst[sb1 + 51] = r1[3];
            if (tid < 32) idT[nxt][tid] = idv;
        }
    }
    __syncthreads();
    // combine 8 waves' D matrices in LDS
    #pragma unroll
    for (int j = 0; j < 8; ++j) {
        int mi = j + hi * 8;               // D layout: vgpr j -> row j (lanes<16) / j+8
        atomicAdd(&sred[mi * 16 + m], dsum[j]);
        atomicAdd(&sred[256 + mi * 16 + m], dssq[j]);
    }
    __syncthreads();
    {
        int mi = tid >> 4, ch = tid & 15;
        int gidx = (n * K_INST + mi) * C_CH + cBase + ch;
        atomicAdd(&sum[gidx], sred[tid]);
        atomicAdd(&ssq[gidx], sred[256 + tid]);
    }
}

// ---------------- scale/bias table ----------------
__global__ void scalebias_k(const float* __restrict__ sum, const float* __restrict__ ssq,
                            const float* __restrict__ cnt, const float* __restrict__ gamma,
                            const float* __restrict__ beta, float* __restrict__ scaleT,
                            float* __restrict__ biasT) {
    int t = blockIdx.x * 256 + threadIdx.x;
    if (t >= N_IMG * K_INST * C_CH) return;
    int n = t >> 10, k = (t >> 6) & 15, c = t & 63;
    float cv = cnt[n * K_INST + k]; cv = cv > 1.f ? cv : 1.f;
    float mean = sum[t] / cv;
    float var  = ssq[t] / cv - mean * mean;
    float s = rsqrtf(var + EPSF) * gamma[c];
    scaleT[t] = s;
    biasT[t]  = beta[c] - mean * s;
}

// ---------------- normalize: out = valid ? x*scale[id] + bias[id] : x ----------------
// x is L2-resident after the stats pass (134MB of 192MB L2): read it last-use/NT
// and stream the output NT so it doesn't evict anything.
__global__ __launch_bounds__(256) void normalize_k(const float* __restrict__ x,
                                                   const signed char* __restrict__ id8,
                                                   const float* __restrict__ scaleT,
                                                   const float* __restrict__ biasT,
                                                   float* __restrict__ out) {
    __shared__ float2 sb[K_INST];          // fused (scale, bias): one b64 gather per pixel
    long long e0 = (long long)blockIdx.x * 4096;      // 4096 elems, within one (n,c) plane
    int n   = (int)(e0 >> 22);                        // C*HW = 2^22
    int c   = (int)((e0 >> 16) & 63);
    int hw0 = (int)(e0 & 65535);
    int tid = threadIdx.x;
    if (tid < K_INST) {
        int gi = (n * K_INST + tid) * C_CH + c;
        sb[tid] = make_float2(scaleT[gi], biasT[gi]);
    }
    __syncthreads();
    const v4f* xp = (const v4f*)(x + ((size_t)(n * C_CH + c) << 16) + hw0);
    v4f*       op = (v4f*)((float*)out + ((size_t)(n * C_CH + c) << 16) + hw0);
    const unsigned int* idp = (const unsigned int*)(id8 + n * HW_SZ + hw0);
    #pragma unroll
    for (int it = 0; it < 4; ++it) {
        int v = it * 256 + tid;
        v4f xv = __builtin_nontemporal_load(&xp[v]);
        unsigned int iv = idp[v];
        v4f ov = xv;
        #pragma unroll
        for (int j = 0; j < 4; ++j) {
            int k = (int)(signed char)((iv >> (8 * j)) & 0xff);
            if (k >= 0) {
                float2 p = sb[k];
                ov[j] = fmaf(xv[j], p.x, p.y);
            }
        }
        __builtin_nontemporal_store(ov, &op[v]);
    }
}

// ---------------- launch ----------------
extern "C" void kernel_launch(void* const* d_in, const int* in_sizes, int n_in,
                              void* d_out, int out_size, void* d_ws, size_t ws_size,
                              hipStream_t stream) {
    const float* x     = (const float*)d_in[0];
    const int*   ids   = (const int*)d_in[1];
    const float* gamma = (const float*)d_in[2];
    const float* beta  = (const float*)d_in[3];
    float* out = (float*)d_out;

    char* ws = (char*)d_ws;
    signed char* id8 = (signed char*)ws;                 // 524288 B
    float* sum    = (float*)(ws + 524288);               // 8192 f32
    float* ssq    = (float*)(ws + 557056);               // 8192 f32
    float* cnt    = (float*)(ws + 589824);               // 128 f32
    float* scaleT = (float*)(ws + 590336);               // 8192 f32
    float* biasT  = (float*)(ws + 623104);               // 8192 f32

    zero_k     <<<65,   256, 0, stream>>>(sum, ssq, cnt);
    prep_k     <<<512,  256, 0, stream>>>(ids, id8, cnt);
    stats_wmma <<<512,  256, 0, stream>>>(x, id8, sum, ssq);
    scalebias_k<<<32,   256, 0, stream>>>(sum, ssq, cnt, gamma, beta, scaleT, biasT);
    normalize_k<<<8192, 256, 0, stream>>>(x, id8, scaleT, biasT, out);
}